// GraphEncoder_89550068122084
// MI455X (gfx1250) — compile-verified
//
#include <hip/hip_runtime.h>
#include <hip/hip_bf16.h>

#define NN 50000
#define EE 800000
#define INDIM 128
#define HIDC 32
#define NHEADS 4
#define LATD 64
#define NG 512
#define NEG_SLOPE 0.2f

typedef __attribute__((ext_vector_type(2))) float v2f;
typedef __attribute__((ext_vector_type(8))) float v8f;

// ---- monotone float<->uint order mapping (for atomic segment-max) ----
__device__ __forceinline__ unsigned f2ord(float f) {
  unsigned u = __float_as_uint(f);
  return (u & 0x80000000u) ? ~u : (u | 0x80000000u);
}
__device__ __forceinline__ float ord2f(unsigned u) {
  return (u & 0x80000000u) ? __uint_as_float(u & 0x7fffffffu)
                           : __uint_as_float(~u);
}

// ---- WMMA f32 GEMM:  H[N,FOUT] = X[N,FIN] @ W[FIN,FOUT] ----
// one 16x16 output tile per wave; 8 waves (wave32) per block share an
// LDS-staged 128x16 W panel. N = 50000 = 3125*16 exactly.
template <int FIN, int FOUT>
__global__ void k_gemm_wmma(const float* __restrict__ X,
                            const float* __restrict__ W,
                            float* __restrict__ H) {
  __shared__ float Wt[FIN * 16];
  const int lane = threadIdx.x & 31;
  const int wave = threadIdx.x >> 5;
  const int n0 = blockIdx.y * 16;

  // stage W[:, n0:n0+16] into LDS (coalesced)
  for (int r = (int)(threadIdx.x >> 4); r < FIN; r += 16)
    Wt[r * 16 + (threadIdx.x & 15)] = W[r * FOUT + n0 + (threadIdx.x & 15)];
  __syncthreads();

  const int mtile = blockIdx.x * 8 + wave;
  if (mtile >= NN / 16) return;          // wave-uniform exit (EXEC stays all-1s)
  const int m0 = mtile * 16;
  const int hi = lane >> 4;              // 0: K={0,1}, 1: K={2,3} (A frag)
  const int nl = lane & 15;              // A row / B column within tile

  v8f c = {};
  const float* xrow = X + (size_t)(m0 + nl) * FIN;
#pragma unroll 8
  for (int k0 = 0; k0 < FIN; k0 += 4) {
    const int ka = k0 + 2 * hi;
    v2f a, b;
    a.x = xrow[ka];
    a.y = xrow[ka + 1];
    b.x = Wt[ka * 16 + nl];
    b.y = Wt[(ka + 1) * 16 + nl];
    c = __builtin_amdgcn_wmma_f32_16x16x4_f32(false, a, false, b,
                                              (short)0, c, false, false);
  }
#pragma unroll
  for (int i = 0; i < 8; ++i) {
    const int row = m0 + i + hi * 8;     // C/D: VGPR i -> M=i (lo), M=i+8 (hi)
    H[(size_t)row * FOUT + n0 + nl] = c[i];
  }
}

// ---- per-(node,head) attention logits ----
__global__ void k_scores(const float* __restrict__ H,
                         const float* __restrict__ a_s,
                         const float* __restrict__ a_d,
                         float* __restrict__ als, float* __restrict__ ald,
                         int heads) {
  const int gid = blockIdx.x * blockDim.x + threadIdx.x;
  if (gid >= NN * heads) return;
  const int n = gid / heads, h = gid % heads;
  const float* hv = H + (size_t)n * heads * HIDC + h * HIDC;
  float ss = 0.f, sd = 0.f;
#pragma unroll
  for (int c = 0; c < HIDC; ++c) {
    const float v = hv[c];
    ss = fmaf(v, a_s[h * HIDC + c], ss);
    sd = fmaf(v, a_d[h * HIDC + c], sd);
  }
  als[gid] = ss;
  ald[gid] = sd;
}

__global__ void k_fill_u32(unsigned* __restrict__ p, unsigned v, int n) {
  const int i = blockIdx.x * blockDim.x + threadIdx.x;
  if (i < n) p[i] = v;
}

__device__ __forceinline__ void edge_sd(int e, const int* __restrict__ es,
                                        const int* __restrict__ ed,
                                        int& s, int& d) {
  if (e < EE) { s = es[e]; d = ed[e]; }
  else        { s = e - EE; d = s; }     // appended self-loops
}

__device__ __forceinline__ float lrelu(float v) {
  return v > 0.f ? v : NEG_SLOPE * v;
}

// ---- segment max over incoming edges ----
__global__ void k_edge_max(const float* __restrict__ als,
                           const float* __restrict__ ald,
                           const int* __restrict__ es, const int* __restrict__ ed,
                           unsigned* __restrict__ mOrd, int heads) {
  const int gid = blockIdx.x * blockDim.x + threadIdx.x;
  if (gid >= (EE + NN) * heads) return;
  const int e = gid / heads, h = gid % heads;
  int s, d; edge_sd(e, es, ed, s, d);
  const float v = lrelu(als[s * heads + h] + ald[d * heads + h]);
  atomicMax(&mOrd[d * heads + h], f2ord(v));
}

// ---- segment sum of exp(e - max) ----
__global__ void k_edge_sum(const float* __restrict__ als,
                           const float* __restrict__ ald,
                           const unsigned* __restrict__ mOrd,
                           float* __restrict__ sSum,
                           const int* __restrict__ es, const int* __restrict__ ed,
                           int heads) {
  const int gid = blockIdx.x * blockDim.x + threadIdx.x;
  if (gid >= (EE + NN) * heads) return;
  const int e = gid / heads, h = gid % heads;
  int s, d; edge_sd(e, es, ed, s, d);
  const float v = lrelu(als[s * heads + h] + ald[d * heads + h]);
  const float ex = __expf(v - ord2f(mOrd[d * heads + h]));
  atomicAdd(&sSum[d * heads + h], ex);
}

// ---- message aggregation: out[dst] += alpha * h[src], one thread per (edge,channel)
__global__ void k_edge_agg(const float* __restrict__ H,
                           const float* __restrict__ als,
                           const float* __restrict__ ald,
                           const unsigned* __restrict__ mOrd,
                           const float* __restrict__ sSum,
                           const int* __restrict__ es, const int* __restrict__ ed,
                           float* __restrict__ out, int heads) {
  const int F = heads * HIDC;
  const int gid = blockIdx.x * blockDim.x + threadIdx.x;
  if (gid >= (EE + NN) * F) return;
  const int f = gid % F;
  const int e = gid / F;
  const int h = f >> 5;                  // f / HIDC
  int s, d; edge_sd(e, es, ed, s, d);
  const float v = lrelu(als[s * heads + h] + ald[d * heads + h]);
  const float alpha =
      __expf(v - ord2f(mOrd[d * heads + h])) / sSum[d * heads + h];
  atomicAdd(&out[(size_t)d * F + f], H[(size_t)s * F + f] * alpha);
}

// ---- bias + ELU, in place ----
__global__ void k_bias_elu(float* __restrict__ h, const float* __restrict__ b,
                           int F) {
  const int gid = blockIdx.x * blockDim.x + threadIdx.x;
  if (gid >= NN * F) return;
  const float v = h[gid] + b[gid % F];
  h[gid] = v > 0.f ? v : __expf(v) - 1.f;
}

// ---- global mean pool (accumulate) ----
__global__ void k_pool(const float* __restrict__ h3,
                       const int* __restrict__ batch,
                       float* __restrict__ pooled, float* __restrict__ cnt) {
  const int gid = blockIdx.x * blockDim.x + threadIdx.x;
  if (gid >= NN * HIDC) return;
  const int n = gid / HIDC, f = gid % HIDC;
  const int g = batch[n];
  atomicAdd(&pooled[g * HIDC + f], h3[gid]);
  if (f == 0) atomicAdd(&cnt[g], 1.f);
}

// ---- mu / logvar heads ----
__global__ void k_head(const float* __restrict__ pooled,
                       const float* __restrict__ cnt,
                       const float* __restrict__ Wmu, const float* __restrict__ bmu,
                       const float* __restrict__ Wlv, const float* __restrict__ blv,
                       float* __restrict__ out) {
  const int gid = blockIdx.x * blockDim.x + threadIdx.x;
  if (gid >= NG * LATD) return;
  const int g = gid / LATD, l = gid % LATD;
  const float inv = 1.f / fmaxf(cnt[g], 1.f);
  float mu = bmu[l], lv = blv[l];
#pragma unroll
  for (int k = 0; k < HIDC; ++k) {
    const float pm = pooled[g * HIDC + k] * inv;
    mu = fmaf(pm, Wmu[k * LATD + l], mu);
    lv = fmaf(pm, Wlv[k * LATD + l], lv);
  }
  out[gid] = mu;
  out[NG * LATD + gid] = lv;
}

// ==================== host-side orchestration ====================
static inline int cdiv(long long a, int b) { return (int)((a + b - 1) / b); }

static void run_layer(const float* Xin, const float* W, const float* a_s,
                      const float* a_d, const float* b, int Fout, int heads,
                      float* hT, float* hO, float* als, float* ald,
                      unsigned* mOrd, float* sSum, const int* es, const int* ed,
                      hipStream_t stream) {
  const dim3 gblk(256);
  const dim3 ggrid(cdiv(NN / 16, 8), Fout / 16);
  if (Fout == 128)
    k_gemm_wmma<128, 128><<<ggrid, gblk, 0, stream>>>(Xin, W, hT);
  else
    k_gemm_wmma<128, 32><<<ggrid, gblk, 0, stream>>>(Xin, W, hT);

  const int nh = NN * heads;
  k_scores<<<cdiv(nh, 256), 256, 0, stream>>>(hT, a_s, a_d, als, ald, heads);
  k_fill_u32<<<cdiv(nh, 256), 256, 0, stream>>>(mOrd, 0u, nh);
  k_fill_u32<<<cdiv(nh, 256), 256, 0, stream>>>((unsigned*)sSum, 0u, nh);
  const int nf = NN * Fout;
  k_fill_u32<<<cdiv(nf, 256), 256, 0, stream>>>((unsigned*)hO, 0u, nf);

  const long long E2 = (long long)(EE + NN);
  k_edge_max<<<cdiv(E2 * heads, 256), 256, 0, stream>>>(als, ald, es, ed, mOrd,
                                                        heads);
  k_edge_sum<<<cdiv(E2 * heads, 256), 256, 0, stream>>>(als, ald, mOrd, sSum,
                                                        es, ed, heads);
  k_edge_agg<<<cdiv(E2 * Fout, 256), 256, 0, stream>>>(hT, als, ald, mOrd, sSum,
                                                       es, ed, hO, heads);
  k_bias_elu<<<cdiv(nf, 256), 256, 0, stream>>>(hO, b, Fout);
}

extern "C" void kernel_launch(void* const* d_in, const int* in_sizes, int n_in,
                              void* d_out, int out_size, void* d_ws,
                              size_t ws_size, hipStream_t stream) {
  const float* x    = (const float*)d_in[0];
  const float* W1   = (const float*)d_in[1];
  const float* as1  = (const float*)d_in[2];
  const float* ad1  = (const float*)d_in[3];
  const float* b1   = (const float*)d_in[4];
  const float* W2   = (const float*)d_in[5];
  const float* as2  = (const float*)d_in[6];
  const float* ad2  = (const float*)d_in[7];
  const float* b2   = (const float*)d_in[8];
  const float* W3   = (const float*)d_in[9];
  const float* as3  = (const float*)d_in[10];
  const float* ad3  = (const float*)d_in[11];
  const float* b3   = (const float*)d_in[12];
  const float* Wmu  = (const float*)d_in[13];
  const float* bmu  = (const float*)d_in[14];
  const float* Wlv  = (const float*)d_in[15];
  const float* blv  = (const float*)d_in[16];
  const int*   eidx = (const int*)d_in[17];
  const int*   batch= (const int*)d_in[18];
  const int* es = eidx;           // edge_index[0]
  const int* ed = eidx + EE;      // edge_index[1]

  float* ws = (float*)d_ws;
  float*    hT     = ws;                        // N*128  transformed features
  float*    hO     = hT + (size_t)NN * 128;     // N*128  aggregated / next input
  float*    als    = hO + (size_t)NN * 128;     // N*4
  float*    ald    = als + (size_t)NN * NHEADS; // N*4
  unsigned* mOrd   = (unsigned*)(ald + (size_t)NN * NHEADS); // N*4
  float*    sSum   = (float*)mOrd + (size_t)NN * NHEADS;     // N*4
  float*    pooled = sSum + (size_t)NN * NHEADS;             // G*32
  float*    cnt    = pooled + (size_t)NG * HIDC;             // G

  // layer 1: x -> hO
  run_layer(x,  W1, as1, ad1, b1, 128, NHEADS, hT, hO, als, ald, mOrd, sSum, es, ed, stream);
  // layer 2: hO -> hO (hT used as intermediate; hO dead after its GEMM read)
  run_layer(hO, W2, as2, ad2, b2, 128, NHEADS, hT, hO, als, ald, mOrd, sSum, es, ed, stream);
  // layer 3 (1 head, 32 ch): hO -> hO[0:N*32)
  run_layer(hO, W3, as3, ad3, b3, 32, 1, hT, hO, als, ald, mOrd, sSum, es, ed, stream);

  // global mean pool + heads
  const int gp = NG * HIDC;
  k_fill_u32<<<cdiv(gp + NG, 256), 256, 0, stream>>>((unsigned*)pooled, 0u, gp + NG);
  k_pool<<<cdiv((long long)NN * HIDC, 256), 256, 0, stream>>>(hO, batch, pooled, cnt);
  k_head<<<cdiv(NG * LATD, 256), 256, 0, stream>>>(pooled, cnt, Wmu, bmu, Wlv, blv,
                                                   (float*)d_out);
}